// PairGeometricEncoder_7387343749846
// MI455X (gfx1250) — compile-verified
//
#include <hip/hip_runtime.h>

// Problem constants (match reference)
#define NB    16
#define BATCH 4
#define NPART 24
#define NSUM  3072

typedef __attribute__((ext_vector_type(2))) float v2f;
typedef __attribute__((ext_vector_type(8))) float v8f;

// ---------------------------------------------------------------------------
// K1: centroids [B,P,3] + per-point part index [B,NS]
// ---------------------------------------------------------------------------
__global__ __launch_bounds__(128) void centroid_kernel(
    const float* __restrict__ part_pcs, const int* __restrict__ n_pcs,
    float* __restrict__ centroids, int* __restrict__ part_idx)
{
    const int blk = blockIdx.x;
    const int b = blk / NPART, p = blk % NPART;
    const int* np = n_pcs + b * NPART;
    int start = 0;
    for (int q = 0; q < p; ++q) start += np[q];
    const int cnt = np[p];

    float sx = 0.f, sy = 0.f, sz = 0.f;
    for (int i = threadIdx.x; i < cnt; i += blockDim.x) {
        const size_t off = (size_t)b * NSUM + start + i;
        const float* pt = part_pcs + off * 3;
        sx += pt[0]; sy += pt[1]; sz += pt[2];
        part_idx[b * NSUM + start + i] = p;
    }
    __shared__ float rx[128], ry[128], rz[128];
    const int t = threadIdx.x;
    rx[t] = sx; ry[t] = sy; rz[t] = sz;
    __syncthreads();
    for (int s = 64; s > 0; s >>= 1) {
        if (t < s) { rx[t] += rx[t+s]; ry[t] += ry[t+s]; rz[t] += rz[t+s]; }
        __syncthreads();
    }
    if (t == 0) {
        const float inv = 1.0f / (float)(cnt > 0 ? cnt : 1);
        float* c = centroids + (b * NPART + p) * 3;
        c[0] = rx[0] * inv; c[1] = ry[0] * inv; c[2] = rz[0] * inv;
    }
}

// ---------------------------------------------------------------------------
// K2: per-batch pair features + (feat @ W + b) via V_WMMA_F32_16X16X4_F32
// One block per batch, 512 threads (16 wave32s). Branch-free WMMA operands.
// ---------------------------------------------------------------------------
__global__ __launch_bounds__(512) void pair_kernel(
    const float* __restrict__ centroids, const int* __restrict__ n_pcs,
    const float* __restrict__ W, const float* __restrict__ bias_p,
    float* __restrict__ pair_bias)
{
    const int b = blockIdx.x;
    const int t = threadIdx.x;

    __shared__ float s_cent[NPART][3];
    __shared__ float s_mask[NPART];
    __shared__ float s_W[32];
    __shared__ float s_vhat[NPART][NPART][3]; // vhat[i][j] = (c_i - c_j)/max(|.|,1e-8)
    __shared__ float s_dist[NPART][NPART];
    __shared__ float s_feat[288][32];         // half of the 576 pairs per pass

    if (t < NPART * 3) ((float*)s_cent)[t] = centroids[b * NPART * 3 + t];
    if (t < NPART)     s_mask[t] = (n_pcs[b * NPART + t] > 0) ? 1.0f : 0.0f;
    if (t < 32)        s_W[t] = W[t];
    __syncthreads();

    for (int idx = t; idx < NPART * NPART; idx += 512) {
        const int i = idx / NPART, j = idx % NPART;
        const float d0 = s_cent[i][0] - s_cent[j][0];
        const float d1 = s_cent[i][1] - s_cent[j][1];
        const float d2 = s_cent[i][2] - s_cent[j][2];
        const float nrm = sqrtf(d0*d0 + d1*d1 + d2*d2);
        s_dist[i][j] = nrm;
        const float inv = 1.0f / fmaxf(nrm, 1e-8f);
        s_vhat[i][j][0] = d0 * inv;
        s_vhat[i][j][1] = d1 * inv;
        s_vhat[i][j][2] = d2 * inv;
    }
    __syncthreads();

    const float bval = bias_p[0];
    const int lane = t & 31;
    const int wv   = t >> 5;

    // Branch-free WMMA operand geometry:
    //   A (f32 16x4): lane<16 holds {K0,K1} of row (lane&15); lane>=16 holds {K2,K3}.
    //   B (4x16): lane<16 holds {B[0][N],B[1][N]}; lane>=16 holds {B[2][N],B[3][N]}.
    //   We want B column N=0 = W chunk, all other columns zero.
    const int   sel     = (lane >> 4) << 1;                  // 0 or 2
    const float col0msk = ((lane & 15) == 0) ? 1.0f : 0.0f;  // N==0 mask

    // B operand chunks are tile-invariant: build once (16 VGPRs).
    v2f bop[8];
    #pragma unroll
    for (int kc = 0; kc < 8; ++kc) {
        const int k0 = kc * 4 + sel;
        bop[kc].x = s_W[k0]     * col0msk;
        bop[kc].y = s_W[k0 + 1] * col0msk;
    }

    for (int pass = 0; pass < 2; ++pass) {
        const int pairBase = pass * 288;

        // Feature build: work item = (pair-in-pass, basis)
        for (int w = t; w < 288 * 16; w += 512) {
            const int pr = w >> 4, base = w & 15;
            const int pidx = pairBase + pr;
            const int i = pidx / NPART, j = pidx % NPART;
            // distance RBF: centers linspace(0,10,16), width 0.625
            const float cd = (10.0f / 15.0f) * (float)base;
            const float zd = (s_dist[i][j] - cd) * 1.6f;
            const float fd = __expf(-0.5f * zd * zd);
            // angle RBF sum over k: cos = vhat[j][i] . vhat[j][k]
            const float ca = -1.0f + (2.0f / 15.0f) * (float)base;
            const float v0 = s_vhat[j][i][0], v1 = s_vhat[j][i][1], v2 = s_vhat[j][i][2];
            float acc = 0.0f;
            #pragma unroll 4
            for (int k = 0; k < NPART; ++k) {
                float dot = v0 * s_vhat[j][k][0] + v1 * s_vhat[j][k][1] + v2 * s_vhat[j][k][2];
                dot = fminf(fmaxf(dot, -1.0f), 1.0f);
                const float za = (dot - ca) * 8.0f;
                acc += s_mask[k] * __expf(-0.5f * za * za);
            }
            s_feat[pr][base]      = fd;
            s_feat[pr][16 + base] = acc;
        }
        __syncthreads();

        // Mat-vec feat[288][32] x W[32] via f32 WMMA: 18 row-tiles x 8 K-chunks.
        for (int tile = wv; tile < 18; tile += 16) {
            const int r0  = tile * 16;
            const int row = r0 + (lane & 15);
            v8f c = {0.f, 0.f, 0.f, 0.f, 0.f, 0.f, 0.f, 0.f};
            #pragma unroll
            for (int kc = 0; kc < 8; ++kc) {
                const int k0 = kc * 4 + sel;
                v2f a;
                a.x = s_feat[row][k0];
                a.y = s_feat[row][k0 + 1];
                c = __builtin_amdgcn_wmma_f32_16x16x4_f32(
                        false, a, false, bop[kc], (short)0, c, false, false);
            }
            // D[M][0]: lane 0 holds M=0..7 in c[0..7]; lane 16 holds M=8..15.
            const int outBase = b * NPART * NPART + pairBase + r0;
            if (lane == 0) {
                #pragma unroll
                for (int m = 0; m < 8; ++m) pair_bias[outBase + m] = c[m] + bval;
            } else if (lane == 16) {
                #pragma unroll
                for (int m = 0; m < 8; ++m) pair_bias[outBase + 8 + m] = c[m] + bval;
            }
        }
        __syncthreads();
    }
}

// ---------------------------------------------------------------------------
// K3: expand columns to point level once: row_vals[b,i,m] = pb[b,i,pi[b,m]]
// ---------------------------------------------------------------------------
__global__ __launch_bounds__(256) void rowvals_kernel(
    const float* __restrict__ pair_bias, const int* __restrict__ part_idx,
    float* __restrict__ row_vals)
{
    const int blk = blockIdx.x;
    const int b = blk / NPART, i = blk % NPART;
    const float* pb = pair_bias + (b * NPART + i) * NPART;
    const int*   pi = part_idx + b * NSUM;
    float* rv = row_vals + (size_t)(b * NPART + i) * NSUM;
    for (int m = threadIdx.x; m < NSUM; m += 256)
        rv[m] = pb[pi[m]];
}

// ---------------------------------------------------------------------------
// K4: bandwidth kernel — out[b,n,:] = row_vals[b, pi[b,n], :]
// float4 coalesced; reads hit L2 (1.2 MB resident), writes stream to HBM.
// ---------------------------------------------------------------------------
__global__ __launch_bounds__(256) void expand_kernel(
    const float* __restrict__ row_vals, const int* __restrict__ part_idx,
    float* __restrict__ out)
{
    const int n = blockIdx.y % NSUM;
    const int b = blockIdx.y / NSUM;
    const int pin = part_idx[b * NSUM + n];
    const float4* src = (const float4*)(row_vals + (size_t)(b * NPART + pin) * NSUM);
    float4* dst = (float4*)(out + ((size_t)b * NSUM + n) * NSUM);
    const int c4 = blockIdx.x * 256 + threadIdx.x; // 0..767 (NSUM/4 float4s)
    dst[c4] = src[c4];
}

// ---------------------------------------------------------------------------
extern "C" void kernel_launch(void* const* d_in, const int* in_sizes, int n_in,
                              void* d_out, int out_size, void* d_ws, size_t ws_size,
                              hipStream_t stream) {
    const float* part_pcs = (const float*)d_in[0]; // [B, NS, 3]
    const int*   n_pcs    = (const int*)  d_in[1]; // [B, P]
    const float* W        = (const float*)d_in[2]; // [32]
    const float* bias     = (const float*)d_in[3]; // scalar
    float* out = (float*)d_out;                    // [B,1,NS,NS]

    char* ws = (char*)d_ws;
    float* centroids = (float*)ws;                           //   1,152 B
    float* pair_bias = (float*)(ws + 4096);                  //   9,216 B
    int*   part_idx  = (int*)  (ws + 4096 + 16384);          //  49,152 B
    float* row_vals  = (float*)(ws + 4096 + 16384 + 65536);  // 1,179,648 B (16B aligned)

    hipLaunchKernelGGL(centroid_kernel, dim3(BATCH * NPART), dim3(128), 0, stream,
                       part_pcs, n_pcs, centroids, part_idx);
    hipLaunchKernelGGL(pair_kernel, dim3(BATCH), dim3(512), 0, stream,
                       centroids, n_pcs, W, bias, pair_bias);
    hipLaunchKernelGGL(rowvals_kernel, dim3(BATCH * NPART), dim3(256), 0, stream,
                       pair_bias, part_idx, row_vals);
    hipLaunchKernelGGL(expand_kernel, dim3(NSUM / 1024, BATCH * NSUM), dim3(256), 0, stream,
                       row_vals, part_idx, out);
}